// BPGNN_74569222193435
// MI455X (gfx1250) — compile-verified
//
#include <hip/hip_runtime.h>
#include <cstddef>

typedef __attribute__((ext_vector_type(2))) float v2f;
typedef __attribute__((ext_vector_type(8))) float v8f;

#define LOG8        2.0794415416798357f   // log(8)
#define A_COUP      0.049787068367863944f // exp(-3)
#define ONE_MINUS_A 0.95021293163213606f  // 1 - exp(-3)
#define LOG_NORM    0.29899997f           // log(1 + 7*exp(-3))

// ---------------------------------------------------------------------------
// Kernel 1: log_b0 = log_softmax(x @ W + b) via V_WMMA_F32_16X16X4_F32.
// One wave per 16-row tile of x; W (256x8) zero-padded to 256x16 in LDS.
// Writes log_b0 (scratch) and log_b (= d_out, iteration state).
// ---------------------------------------------------------------------------
__global__ __launch_bounds__(256) void bp_gemm_lsm(
    const float* __restrict__ x, const float* __restrict__ W,
    const float* __restrict__ bias, float* __restrict__ log_b0,
    float* __restrict__ log_b, int n_nodes) {
  __shared__ float Wlds[256 * 16];   // 16 KB: W padded to 16 cols
  __shared__ float Dlds[8][16 * 16]; // 8 KB: per-wave 16x16 D tile

  const int tid = threadIdx.x;
  {
    // 256 threads, each fills one k-row of the padded B matrix
    const int k = tid;
#pragma unroll
    for (int n = 0; n < 16; ++n)
      Wlds[k * 16 + n] = (n < 8) ? W[k * 8 + n] : 0.0f;
  }
  __syncthreads();

  const int wave = tid >> 5;
  const int lane = tid & 31;
  const int half = lane >> 4;   // K sub-block select per ISA f32 A/B layout
  const int r    = lane & 15;   // M index for A, N index for B/D
  const int tile = blockIdx.x * 8 + wave;
  const int row0 = tile * 16;

  if (row0 < n_nodes) {
    v8f acc = {};
    const float* xrow = x + (size_t)(row0 + r) * 256;
#pragma unroll 4
    for (int k = 0; k < 256; k += 4) {
      const int ka = k + 2 * half;
      v2f a, b;
      // A 16x4 f32: VGPR0 = K(2*half), VGPR1 = K(2*half+1), lanes carry M
      a.x = xrow[ka];
      a.y = xrow[ka + 1];
      // B 4x16 f32: dual layout, lanes carry N
      b.x = Wlds[ka * 16 + r];
      b.y = Wlds[(ka + 1) * 16 + r];
      acc = __builtin_amdgcn_wmma_f32_16x16x4_f32(
          false, a, false, b, (short)0, acc, false, false);
    }
    // D 16x16 f32: VGPR v holds row M = v + 8*half, lane%16 = N
#pragma unroll
    for (int v = 0; v < 8; ++v)
      Dlds[wave][(v + 8 * half) * 16 + r] = acc[v];
  }
  __syncthreads();

  // lanes 0..15 of each wave: log-softmax of one row (classes 0..7)
  if (row0 < n_nodes && lane < 16) {
    const int row = row0 + lane; // N is a multiple of 16, all rows valid
    float z[8];
    float mx = -1e30f;
#pragma unroll
    for (int c = 0; c < 8; ++c) {
      z[c] = Dlds[wave][lane * 16 + c] + bias[c];
      mx = fmaxf(mx, z[c]);
    }
    float s = 0.0f;
#pragma unroll
    for (int c = 0; c < 8; ++c) s += __expf(z[c] - mx);
    const float lse = mx + __logf(s);
    float4 o0, o1;
    o0.x = z[0] - lse; o0.y = z[1] - lse; o0.z = z[2] - lse; o0.w = z[3] - lse;
    o1.x = z[4] - lse; o1.y = z[5] - lse; o1.z = z[6] - lse; o1.w = z[7] - lse;
    float4* p0 = (float4*)(log_b0 + (size_t)row * 8);
    float4* p1 = (float4*)(log_b  + (size_t)row * 8);
    p0[0] = o0; p0[1] = o1;
    p1[0] = o0; p1[1] = o1;
  }
}

// ---------------------------------------------------------------------------
// Kernel 2: per-edge message. Uses the closed form for the 0/-3 coupling:
//   msg[c] = log((1-a)*e^{t[c]-m} + a*S) - (log S + log(1+7a)),  a = e^-3
// One thread per edge; rows move as float4 pairs; scatter via f32 atomics.
// ---------------------------------------------------------------------------
template <bool FIRST>
__global__ __launch_bounds__(256) void bp_edge(
    const float* __restrict__ log_b, const float* __restrict__ msg_prev,
    float* __restrict__ msg_new, float* __restrict__ agg,
    const int* __restrict__ src, const int* __restrict__ dst,
    const int* __restrict__ rv, int n_edges) {
  const int e = blockIdx.x * blockDim.x + threadIdx.x;
  if (e >= n_edges) return;

  const int s = src[e];
  const int d = dst[e];

  const float4* xb = (const float4*)(log_b + (size_t)s * 8);
  const float4 x0 = xb[0], x1 = xb[1];
  float t[8] = {x0.x, x0.y, x0.z, x0.w, x1.x, x1.y, x1.z, x1.w};

  if (FIRST) {
#pragma unroll
    for (int c = 0; c < 8; ++c) t[c] += LOG8; // msg0 = -log(8) uniformly
  } else {
    const int rve = rv[e];
    const float4* mb = (const float4*)(msg_prev + (size_t)rve * 8);
    const float4 m0 = mb[0], m1 = mb[1];
    const float mm[8] = {m0.x, m0.y, m0.z, m0.w, m1.x, m1.y, m1.z, m1.w};
#pragma unroll
    for (int c = 0; c < 8; ++c) t[c] -= mm[c];
  }

  float mx = t[0];
#pragma unroll
  for (int c = 1; c < 8; ++c) mx = fmaxf(mx, t[c]);
  float ex[8], S = 0.0f;
#pragma unroll
  for (int c = 0; c < 8; ++c) {
    ex[c] = __expf(t[c] - mx);
    S += ex[c];
  }
  const float norm = __logf(S) + LOG_NORM;
  float msg[8];
#pragma unroll
  for (int c = 0; c < 8; ++c)
    msg[c] = __logf(ONE_MINUS_A * ex[c] + A_COUP * S) - norm;

  float4 w0, w1;
  w0.x = msg[0]; w0.y = msg[1]; w0.z = msg[2]; w0.w = msg[3];
  w1.x = msg[4]; w1.y = msg[5]; w1.z = msg[6]; w1.w = msg[7];
  float4* mo = (float4*)(msg_new + (size_t)e * 8);
  mo[0] = w0; mo[1] = w1;

  float* arow = agg + (size_t)d * 8;
#pragma unroll
  for (int c = 0; c < 8; ++c) unsafeAtomicAdd(arow + c, msg[c]);
}

// ---------------------------------------------------------------------------
// Kernel 3: node update log_b = log_normalize(agg + log_b0); rezero agg.
// ---------------------------------------------------------------------------
__global__ __launch_bounds__(256) void bp_node(
    const float* __restrict__ log_b0, float* __restrict__ agg,
    float* __restrict__ log_b, int n_nodes) {
  const int n = blockIdx.x * blockDim.x + threadIdx.x;
  if (n >= n_nodes) return;
  const size_t off = (size_t)n * 8;

  float4* ap = (float4*)(agg + off);
  const float4* bp = (const float4*)(log_b0 + off);
  const float4 a0 = ap[0], a1 = ap[1];
  const float4 b0 = bp[0], b1 = bp[1];
  float z[8] = {a0.x + b0.x, a0.y + b0.y, a0.z + b0.z, a0.w + b0.w,
                a1.x + b1.x, a1.y + b1.y, a1.z + b1.z, a1.w + b1.w};

  const float4 zero = {0.f, 0.f, 0.f, 0.f};
  ap[0] = zero; ap[1] = zero; // ready for next iteration's atomics

  float mx = z[0];
#pragma unroll
  for (int c = 1; c < 8; ++c) mx = fmaxf(mx, z[c]);
  float s = 0.0f;
#pragma unroll
  for (int c = 0; c < 8; ++c) s += __expf(z[c] - mx);
  const float lse = mx + __logf(s);

  float4 o0, o1;
  o0.x = z[0] - lse; o0.y = z[1] - lse; o0.z = z[2] - lse; o0.w = z[3] - lse;
  o1.x = z[4] - lse; o1.y = z[5] - lse; o1.z = z[6] - lse; o1.w = z[7] - lse;
  float4* op = (float4*)(log_b + off);
  op[0] = o0; op[1] = o1;
}

__global__ __launch_bounds__(256) void bp_zero(float* __restrict__ p, int n) {
  const int i = blockIdx.x * blockDim.x + threadIdx.x;
  if (i < n) p[i] = 0.0f;
}

extern "C" void kernel_launch(void* const* d_in, const int* in_sizes, int n_in,
                              void* d_out, int out_size, void* d_ws,
                              size_t ws_size, hipStream_t stream) {
  const float* x    = (const float*)d_in[0];
  const float* W    = (const float*)d_in[1];
  const float* bias = (const float*)d_in[2];
  const int* eidx   = (const int*)d_in[3]; // [2,E] flat (JAX int64 -> i32 under default x64-off)
  const int* rv     = (const int*)d_in[4];

  const int n_nodes = in_sizes[0] / 256; // DIN = 256
  const int n_edges = in_sizes[4];
  const int* src = eidx;
  const int* dst = eidx + n_edges;

  float* out = (float*)d_out; // log_b lives here across iterations
  float* ws  = (float*)d_ws;
  float* log_b0 = ws;                               // n_nodes*8 floats
  float* agg    = log_b0 + (size_t)n_nodes * 8;     // n_nodes*8 floats
  float* msgA   = agg    + (size_t)n_nodes * 8;     // n_edges*8 floats
  float* msgB   = msgA   + (size_t)n_edges * 8;     // n_edges*8 floats

  const int nf = n_nodes * 8;
  bp_zero<<<(nf + 255) / 256, 256, 0, stream>>>(agg, nf);

  const int n_tiles = (n_nodes + 15) / 16;
  bp_gemm_lsm<<<(n_tiles + 7) / 8, 256, 0, stream>>>(x, W, bias, log_b0, out,
                                                     n_nodes);

  const int eb = (n_edges + 255) / 256;
  const int nb = (n_nodes + 255) / 256;

  // step 0: uniform prior messages (-log 8), write msgA
  bp_edge<true><<<eb, 256, 0, stream>>>(out, nullptr, msgA, agg, src, dst, rv,
                                        n_edges);
  bp_node<<<nb, 256, 0, stream>>>(log_b0, agg, out, n_nodes);

  // steps 1..4: double-buffered messages
  float* prev = msgA;
  float* cur  = msgB;
  for (int step = 1; step < 5; ++step) {
    bp_edge<false><<<eb, 256, 0, stream>>>(out, prev, cur, agg, src, dst, rv,
                                           n_edges);
    bp_node<<<nb, 256, 0, stream>>>(log_b0, agg, out, n_nodes);
    float* tmp = prev; prev = cur; cur = tmp;
  }
}